// SymmetricContraction_10033043604100
// MI455X (gfx1250) — compile-verified
//
#include <hip/hip_runtime.h>

typedef __attribute__((ext_vector_type(2))) float v2f;
typedef __attribute__((ext_vector_type(8))) float v8f;

#define CN 128   // channels
#define DN 4     // irreps dim
#define EN 10    // elements
#define BT 16    // b-rows per block (WMMA M)
#define CT 64    // channels per block
#define NK 9     // total k-columns across all (order, irrep) tensors
#define NT (NK * CT)      // 576 LDS W columns per block
#define NTILE (NT / 16)   // 36 WMMA N-tiles

// symmetric monomial tables, lexicographic p<=q<=j over D=4
__device__ __constant__ int   TRI_P[20] = {0,0,0,0,0,0,0,0,0,0,1,1,1,1,1,1,2,2,2,3};
__device__ __constant__ int   TRI_Q[20] = {0,0,0,0,1,1,1,2,2,3,1,1,1,2,2,3,2,2,3,3};
__device__ __constant__ int   TRI_J[20] = {0,1,2,3,1,2,3,2,3,3,1,2,3,2,3,3,2,3,3,3};
__device__ __constant__ float TRI_M[20] = {1,3,3,3,3,6,6,3,6,3,1,3,3,3,6,3,1,3,3,1};
__device__ __constant__ int   PR_P[10]  = {0,0,0,0,1,1,1,2,2,3};
__device__ __constant__ int   PR_Q[10]  = {0,1,2,3,1,2,3,2,3,3};
__device__ __constant__ float PR_M[10]  = {1,2,2,2,1,2,2,1,2,1};

__device__ __forceinline__ float dotN(const float* cu, const float* m, int n4) {
  float s = 0.f;
#pragma unroll
  for (int q = 0; q < n4; ++q) {
    float4 cv = ((const float4*)cu)[q];
    s = fmaf(cv.x, m[4*q+0], s);
    s = fmaf(cv.y, m[4*q+1], s);
    s = fmaf(cv.z, m[4*q+2], s);
    s = fmaf(cv.w, m[4*q+3], s);
  }
  return s;
}

__global__ __launch_bounds__(256) void symcon_kernel(
    const float* __restrict__ x,  const float* __restrict__ y,
    const float* __restrict__ u3_0e, const float* __restrict__ w3_0e,
    const float* __restrict__ u3_1o, const float* __restrict__ w3_1o,
    const float* __restrict__ u2_0e, const float* __restrict__ w2_0e,
    const float* __restrict__ u2_1o, const float* __restrict__ w2_1o,
    const float* __restrict__ u1_0e, const float* __restrict__ w1_0e,
    const float* __restrict__ u1_1o, const float* __restrict__ w1_1o,
    float* __restrict__ out) {
  __shared__ __align__(16) float s_W[BT * NT];      // W[b_local][kidx][c_local]
  __shared__ __align__(16) float s_cu3_0e[2 * 20];
  __shared__ __align__(16) float s_cu3_1o[6 * 20];  // [k][i][t]
  __shared__ __align__(16) float s_cu2_0e[2 * 12];  // padded stride 12
  __shared__ __align__(16) float s_cu2_1o[3 * 12];  // [i][t] padded
  __shared__ __align__(16) float s_cu1_0e[4];
  __shared__ __align__(16) float s_cu1_1o[12];      // [i][j]

  const int tid  = threadIdx.x;
  // Force the wave id into an SGPR so the tile loop + kidx switch compile as
  // scalar control flow (EXEC stays all-ones through the WMMAs).
  const int wave = __builtin_amdgcn_readfirstlane(tid >> 5);
  const int lane = tid & 31;
  const int half = lane >> 4;      // WMMA lane half
  const int l16  = lane & 15;
  const int b0   = blockIdx.x * BT;
  const int c0   = blockIdx.y * CT;

  // ---------------- Stage 1: W = y @ (w/mul) via V_WMMA_F32_16X16X4_F32 -----
  // A operand (16x4 f32): VGPR0 lanes0-15 K=kb+0, lanes16-31 K=kb+2;
  //                       VGPR1 lanes0-15 K=kb+1, lanes16-31 K=kb+3.
  v2f Aop[3];
#pragma unroll
  for (int kc = 0; kc < 3; ++kc) {
    int kb = kc * 4;
    int ex = kb + 2 * half, ey = ex + 1;
    Aop[kc].x = (ex < EN) ? y[(b0 + l16) * EN + ex] : 0.0f;
    Aop[kc].y = (ey < EN) ? y[(b0 + l16) * EN + ey] : 0.0f;
  }

  for (int t = wave; t < NTILE; t += 8) {   // t is scalar (SGPR) -> s_cbranch
    const int n0   = t * 16;
    const int kidx = n0 / CT;               // scalar, uniform over the tile
    const int c    = c0 + (n0 % CT) + l16;
    const float* wp; int kk; float sc;
    switch (kidx) {                         // scalar branches now
      case 0:  wp = w3_0e; kk = 0; sc = 0.5f; break;
      case 1:  wp = w3_0e; kk = 1; sc = 0.5f; break;
      case 2:  wp = w3_1o; kk = 0; sc = 0.5f; break;
      case 3:  wp = w3_1o; kk = 1; sc = 0.5f; break;
      case 4:  wp = w2_0e; kk = 0; sc = 0.5f; break;
      case 5:  wp = w2_0e; kk = 1; sc = 0.5f; break;
      case 6:  wp = w2_1o; kk = 0; sc = 1.0f; break;
      case 7:  wp = w1_0e; kk = 0; sc = 1.0f; break;
      default: wp = w1_1o; kk = 0; sc = 1.0f; break;
    }
    v8f acc = {};
#pragma unroll
    for (int kc = 0; kc < 3; ++kc) {
      int kb = kc * 4;
      int ex = kb + 2 * half, ey = ex + 1;
      v2f Bop;
      Bop.x = (ex < EN) ? wp[(kk * EN + ex) * CN + c] * sc : 0.0f;
      Bop.y = (ey < EN) ? wp[(kk * EN + ey) * CN + c] * sc : 0.0f;
      acc = __builtin_amdgcn_wmma_f32_16x16x4_f32(
          false, Aop[kc], false, Bop, (short)0, acc, false, false);
    }
    // D layout: VGPR r, lanes0-15 -> M=r, lanes16-31 -> M=8+r; N = lane&15
#pragma unroll
    for (int r = 0; r < 8; ++r)
      s_W[(r + 8 * half) * NT + n0 + l16] = acc[r];
  }

  // ---------------- Stage 2: fold symmetric u tensors -> monomial coeffs ----
  if (tid == 255) {  // zero the 16B-stride pads (only writer of these slots)
    s_cu2_0e[10] = s_cu2_0e[11] = s_cu2_0e[22] = s_cu2_0e[23] = 0.f;
    s_cu2_1o[10] = s_cu2_1o[11] = s_cu2_1o[22] = s_cu2_1o[23] = 0.f;
    s_cu2_1o[34] = s_cu2_1o[35] = 0.f;
  }
  {
    int t = tid;
    if (t < 40) {                       // u3_0e: (p,q,j,k,1)
      int k = t / 20, m = t % 20;
      s_cu3_0e[t] =
          u3_0e[((TRI_P[m]*DN + TRI_Q[m])*DN + TRI_J[m])*2 + k] * TRI_M[m];
    } else if (t < 160) {               // u3_1o: (p,q,j,k,3)
      int idx = t - 40, k = idx / 60, r = idx % 60, i = r / 20, m = r % 20;
      s_cu3_1o[idx] =
          u3_1o[(((TRI_P[m]*DN + TRI_Q[m])*DN + TRI_J[m])*2 + k)*3 + i] * TRI_M[m];
    } else if (t < 180) {               // u2_0e: (p,q,k,1)
      int idx = t - 160, k = idx / 10, m = idx % 10;
      s_cu2_0e[k*12 + m] = u2_0e[(PR_P[m]*DN + PR_Q[m])*2 + k] * PR_M[m];
    } else if (t < 210) {               // u2_1o: (p,q,1,3)
      int idx = t - 180, i = idx / 10, m = idx % 10;
      s_cu2_1o[i*12 + m] = u2_1o[(PR_P[m]*DN + PR_Q[m])*3 + i] * PR_M[m];
    } else if (t < 214) {               // u1_0e: (j,1,1)
      int j = t - 210;
      s_cu1_0e[j] = u1_0e[j];
    } else if (t < 226) {               // u1_1o: (j,1,3) -> [i][j]
      int idx = t - 214, i = idx >> 2, j = idx & 3;
      s_cu1_1o[i*4 + j] = u1_1o[j*3 + i];
    }
  }
  __syncthreads();

  // ---------------- Stage 3: per-pair polynomial evaluation -----------------
  for (int p = tid; p < BT * CT; p += 256) {
    const int bl = p >> 6, cl = p & (CT - 1);
    const int b = b0 + bl, c = c0 + cl;
    const float4 x4 = *(const float4*)(x + ((size_t)b * CN + c) * DN);
    float xv[4] = {x4.x, x4.y, x4.z, x4.w};

    float P2[12], P3[20];
    {
      int id2 = 0, id3 = 0;
#pragma unroll
      for (int pp = 0; pp < 4; ++pp)
#pragma unroll
        for (int q = pp; q < 4; ++q) {
          float xpq = xv[pp] * xv[q];
          P2[id2++] = xpq;
#pragma unroll
          for (int j = q; j < 4; ++j) P3[id3++] = xpq * xv[j];
        }
    }
    P2[10] = 0.f; P2[11] = 0.f;

    float Wv[9];
#pragma unroll
    for (int k = 0; k < 9; ++k) Wv[k] = s_W[bl * NT + k * CT + cl];

    float o0 = Wv[0] * dotN(s_cu3_0e,      P3, 5)
             + Wv[1] * dotN(s_cu3_0e + 20, P3, 5)
             + Wv[4] * dotN(s_cu2_0e,      P2, 3)
             + Wv[5] * dotN(s_cu2_0e + 12, P2, 3)
             + Wv[7] * dotN(s_cu1_0e,      xv, 1);
    float4 o;
    o.x = o0;
    float oi[3];
#pragma unroll
    for (int i = 0; i < 3; ++i) {
      oi[i] = Wv[2] * dotN(s_cu3_1o + i * 20,        P3, 5)
            + Wv[3] * dotN(s_cu3_1o + (3 + i) * 20,  P3, 5)
            + Wv[6] * dotN(s_cu2_1o + i * 12,        P2, 3)
            + Wv[8] * dotN(s_cu1_1o + i * 4,         xv, 1);
    }
    o.y = oi[0]; o.z = oi[1]; o.w = oi[2];
    *(float4*)(out + ((size_t)b * CN + c) * 4) = o;
  }
}

extern "C" void kernel_launch(void* const* d_in, const int* in_sizes, int n_in,
                              void* d_out, int out_size, void* d_ws, size_t ws_size,
                              hipStream_t stream) {
  const float* x     = (const float*)d_in[0];
  const float* y     = (const float*)d_in[1];
  const float* u3_0e = (const float*)d_in[2];
  const float* w3_0e = (const float*)d_in[3];
  const float* u3_1o = (const float*)d_in[4];
  const float* w3_1o = (const float*)d_in[5];
  const float* u2_0e = (const float*)d_in[6];
  const float* w2_0e = (const float*)d_in[7];
  const float* u2_1o = (const float*)d_in[8];
  const float* w2_1o = (const float*)d_in[9];
  const float* u1_0e = (const float*)d_in[10];
  const float* w1_0e = (const float*)d_in[11];
  const float* u1_1o = (const float*)d_in[12];
  const float* w1_1o = (const float*)d_in[13];
  float* out = (float*)d_out;

  const int B = in_sizes[0] / (CN * DN);  // 16384
  dim3 grid(B / BT, CN / CT);
  symcon_kernel<<<grid, 256, 0, stream>>>(
      x, y, u3_0e, w3_0e, u3_1o, w3_1o, u2_0e, w2_0e, u2_1o, w2_1o,
      u1_0e, w1_0e, u1_1o, w1_1o, out);
}